// GNNModel_4672924418728
// MI455X (gfx1250) — compile-verified
//
#include <hip/hip_runtime.h>

#define N_NODES  10000
#define N_FEAT   256
#define N_GRAPHS 64

typedef __attribute__((ext_vector_type(2))) float v2f;
typedef __attribute__((ext_vector_type(8))) float v8f;

// ---------------------------------------------------------------- init
__global__ void k_init(float* deg, float* aggx, float* agg2, float* gsum, float* gcnt) {
    int t = blockIdx.x * blockDim.x + threadIdx.x;
    if (t < N_NODES)            { deg[t] = 1.0f; aggx[t] = 0.0f; }  // deg starts at 1 (self loop)
    if (t < N_NODES * N_FEAT)   agg2[t] = 0.0f;
    if (t < N_GRAPHS * N_FEAT)  gsum[t] = 0.0f;
    if (t < N_GRAPHS)           gcnt[t] = 0.0f;
}

// ---------------------------------------------------------------- degrees
__global__ void k_deg(const int* __restrict__ dst, int E, float* deg) {
    int e = blockIdx.x * blockDim.x + threadIdx.x;
    if (e < E) atomicAdd(&deg[dst[e]], 1.0f);
}

__global__ void k_dinv(float* deg) {   // deg >= 1 always (self loops)
    int i = blockIdx.x * blockDim.x + threadIdx.x;
    if (i < N_NODES) deg[i] = rsqrtf(deg[i]);
}

// -------------------------------------------------- layer-1 scalar aggregation: aggx = Â x
__global__ void k_aggx(const int* __restrict__ src, const int* __restrict__ dst, int E,
                       const float* __restrict__ x, const float* __restrict__ dinv,
                       float* aggx) {
    int e = blockIdx.x * blockDim.x + threadIdx.x;
    if (e >= E) return;
    int s = src[e], d = dst[e];
    atomicAdd(&aggx[d], dinv[s] * dinv[d] * x[s]);
}

// -------------------------------------------------- h1 = relu(aggx_full * W1 + b1), [N,256]
__global__ void __launch_bounds__(256) k_h1(const float* __restrict__ aggx,
                                            const float* __restrict__ x,
                                            const float* __restrict__ dinv,
                                            const float* __restrict__ W1,
                                            const float* __restrict__ b1,
                                            float* __restrict__ h1) {
    int t = blockIdx.x * blockDim.x + threadIdx.x;
    if (t >= N_NODES * N_FEAT) return;
    int i = t >> 8, j = t & 255;
    float di = dinv[i];
    float a  = aggx[i] + di * di * x[i];          // include self-loop term
    float v  = fmaf(a, W1[j], b1[j]);
    h1[t] = v > 0.0f ? v : 0.0f;
}

// -------------------------------------------------- agg2 += Â h1 (edge scatter, one wave32/edge)
__global__ void __launch_bounds__(256) k_scatter(const int* __restrict__ src,
                                                 const int* __restrict__ dst, int E,
                                                 const float* __restrict__ dinv,
                                                 const float* __restrict__ h1,
                                                 float* __restrict__ agg2) {
    int wave = threadIdx.x >> 5;
    int lane = threadIdx.x & 31;
    int e = blockIdx.x * 8 + wave;                // 8 waves per 256-thread block
    if (e >= E) return;                           // uniform per wave
    int s = src[e], d = dst[e];
    float nrm = dinv[s] * dinv[d];
    const float4* hs = (const float4*)(h1 + (size_t)s * N_FEAT + lane * 8);
    float4 p0 = hs[0], p1 = hs[1];
    float* ad = agg2 + (size_t)d * N_FEAT + lane * 8;
    atomicAdd(ad + 0, p0.x * nrm); atomicAdd(ad + 1, p0.y * nrm);
    atomicAdd(ad + 2, p0.z * nrm); atomicAdd(ad + 3, p0.w * nrm);
    atomicAdd(ad + 4, p1.x * nrm); atomicAdd(ad + 5, p1.y * nrm);
    atomicAdd(ad + 6, p1.z * nrm); atomicAdd(ad + 7, p1.w * nrm);
}

// -------------------------------------------------- agg2 += dinv^2 * h1  (self loops)
__global__ void __launch_bounds__(256) k_selfloop(const float* __restrict__ dinv,
                                                  const float* __restrict__ h1,
                                                  float* __restrict__ agg2) {
    int t = blockIdx.x * blockDim.x + threadIdx.x;
    if (t >= N_NODES * N_FEAT) return;
    int i = t >> 8;
    float di = dinv[i];
    agg2[t] += di * di * h1[t];
}

// -------------------------------------------------- h2 = relu(agg2 @ W2 + b2) via f32 WMMA
// One wave per 16x16 C tile; K-loop of 64 steps of V_WMMA_F32_16X16X4_F32 (exact f32).
// A 16x4 layout : lanes 0-15 -> K=k0,k0+1 ; lanes 16-31 -> K=k0+2,k0+3 (2 VGPRs)
// B 4x16 layout : VGPR0 lanes 0-15 = row k0, lanes 16-31 = row k0+2; VGPR1 = rows k0+1/k0+3
// C 16x16 layout: VGPR r -> rows r (lanes 0-15) and r+8 (lanes 16-31)
__global__ void __launch_bounds__(256) k_gemm(const float* __restrict__ A,
                                              const float* __restrict__ W2,
                                              const float* __restrict__ b2,
                                              float* __restrict__ h2) {
    int lane   = threadIdx.x & 31;
    int wave   = threadIdx.x >> 5;
    int m0     = blockIdx.x * 16;                        // 625 row tiles (10000 = 625*16 exact)
    int n0     = blockIdx.y * 128 + wave * 16;           // 8 waves cover 128 cols
    int laneM  = lane & 15;
    int laneHi = lane >> 4;

    const float* arow = A  + (size_t)(m0 + laneM) * N_FEAT;
    const float* bcol = W2 + (n0 + laneM);

    v8f c = {};
    for (int k0 = 0; k0 < N_FEAT; k0 += 4) {
        int ka = k0 + 2 * laneHi;
        v2f a, b;
        a.x = arow[ka];
        a.y = arow[ka + 1];
        b.x = bcol[(size_t)ka * N_FEAT];
        b.y = bcol[(size_t)(ka + 1) * N_FEAT];
        c = __builtin_amdgcn_wmma_f32_16x16x4_f32(
                /*neg_a=*/false, a, /*neg_b=*/false, b,
                /*c_mod=*/(short)0, c, /*reuse_a=*/false, /*reuse_b=*/false);
    }

    int   col  = n0 + laneM;
    float bias = b2[col];
#pragma unroll
    for (int r = 0; r < 8; ++r) {
        int row = m0 + r + 8 * laneHi;
        float v = c[r] + bias;
        h2[(size_t)row * N_FEAT + col] = v > 0.0f ? v : 0.0f;
    }
}

// -------------------------------------------------- mean pool accumulation
__global__ void __launch_bounds__(256) k_pool(const float* __restrict__ h2,
                                              const int* __restrict__ batch,
                                              float* __restrict__ gsum,
                                              float* __restrict__ gcnt) {
    int i = blockIdx.x;
    int j = threadIdx.x;
    int b = batch[i];
    atomicAdd(&gsum[b * N_FEAT + j], h2[(size_t)i * N_FEAT + j]);
    if (j == 0) atomicAdd(&gcnt[b], 1.0f);
}

// -------------------------------------------------- tiny MLP head, one block per graph
__global__ void __launch_bounds__(256) k_mlp(const float* __restrict__ gsum,
                                             const float* __restrict__ gcnt,
                                             const float* __restrict__ W3, const float* __restrict__ b3,
                                             const float* __restrict__ W4, const float* __restrict__ b4,
                                             const float* __restrict__ W5, const float* __restrict__ b5,
                                             const float* __restrict__ W6, const float* __restrict__ b6,
                                             const float* __restrict__ W7, const float* __restrict__ b7,
                                             float* __restrict__ y) {
    __shared__ float buf0[256];
    __shared__ float buf1[256];
    int g = blockIdx.x, t = threadIdx.x;

    float cnt = gcnt[g];
    cnt = cnt > 1.0f ? cnt : 1.0f;
    buf0[t] = gsum[g * N_FEAT + t] / cnt;
    __syncthreads();

    if (t < 128) {                                   // 256 -> 128
        float s = b3[t];
        for (int k = 0; k < 256; ++k) s = fmaf(buf0[k], W3[k * 128 + t], s);
        buf1[t] = s > 0.0f ? s : 0.0f;
    }
    __syncthreads();
    if (t < 128) {                                   // 128 -> 128
        float s = b4[t];
        for (int k = 0; k < 128; ++k) s = fmaf(buf1[k], W4[k * 128 + t], s);
        buf0[t] = s > 0.0f ? s : 0.0f;
    }
    __syncthreads();
    if (t < 64) {                                    // 128 -> 64
        float s = b5[t];
        for (int k = 0; k < 128; ++k) s = fmaf(buf0[k], W5[k * 64 + t], s);
        buf1[t] = s > 0.0f ? s : 0.0f;
    }
    __syncthreads();
    if (t < 32) {                                    // 64 -> 32
        float s = b6[t];
        for (int k = 0; k < 64; ++k) s = fmaf(buf1[k], W6[k * 32 + t], s);
        buf0[t] = s > 0.0f ? s : 0.0f;
    }
    __syncthreads();
    if (t == 0) {                                    // 32 -> 1
        float s = b7[0];
        for (int k = 0; k < 32; ++k) s = fmaf(buf0[k], W7[k], s);
        y[g] = s;
    }
}

// ================================================================ launch
extern "C" void kernel_launch(void* const* d_in, const int* in_sizes, int n_in,
                              void* d_out, int out_size, void* d_ws, size_t ws_size,
                              hipStream_t stream) {
    const float* x     = (const float*)d_in[0];
    const int*   ei    = (const int*)  d_in[1];
    const int*   batch = (const int*)  d_in[2];
    const float* W1 = (const float*)d_in[3];  const float* b1 = (const float*)d_in[4];
    const float* W2 = (const float*)d_in[5];  const float* b2 = (const float*)d_in[6];
    const float* W3 = (const float*)d_in[7];  const float* b3 = (const float*)d_in[8];
    const float* W4 = (const float*)d_in[9];  const float* b4 = (const float*)d_in[10];
    const float* W5 = (const float*)d_in[11]; const float* b5 = (const float*)d_in[12];
    const float* W6 = (const float*)d_in[13]; const float* b6 = (const float*)d_in[14];
    const float* W7 = (const float*)d_in[15]; const float* b7 = (const float*)d_in[16];

    const int E = in_sizes[1] / 2;
    const int* src = ei;
    const int* dst = ei + E;

    // Workspace layout (all f32). ~29.5 MB total; fits the MI455X 192MB L2 entirely.
    float* ws   = (float*)d_ws;
    float* deg  = ws;                               // [N]  degree -> dinv (in place)
    float* aggx = deg  + N_NODES;                   // [N]  Â·x scalar aggregation
    float* h1   = aggx + N_NODES;                   // [N,256]
    float* agg2 = h1   + (size_t)N_NODES * N_FEAT;  // [N,256]
    float* h2   = agg2 + (size_t)N_NODES * N_FEAT;  // [N,256]
    float* gsum = h2   + (size_t)N_NODES * N_FEAT;  // [G,256]
    float* gcnt = gsum + N_GRAPHS * N_FEAT;         // [G]

    size_t need = (size_t)(2 * N_NODES + 3 * N_NODES * N_FEAT
                           + N_GRAPHS * N_FEAT + N_GRAPHS) * sizeof(float);
    if (ws_size < need) return;                     // insufficient scratch: bail

    const int NT = N_NODES * N_FEAT;                // 2.56M

    k_init    <<<(NT + 255) / 256, 256, 0, stream>>>(deg, aggx, agg2, gsum, gcnt);
    k_deg     <<<(E + 255) / 256, 256, 0, stream>>>(dst, E, deg);
    k_dinv    <<<(N_NODES + 255) / 256, 256, 0, stream>>>(deg);
    k_aggx    <<<(E + 255) / 256, 256, 0, stream>>>(src, dst, E, x, deg, aggx);
    k_h1      <<<(NT + 255) / 256, 256, 0, stream>>>(aggx, x, deg, W1, b1, h1);
    k_scatter <<<(E + 7) / 8, 256, 0, stream>>>(src, dst, E, deg, h1, agg2);
    k_selfloop<<<(NT + 255) / 256, 256, 0, stream>>>(deg, h1, agg2);

    dim3 ggrid(N_NODES / 16, 2);                    // 625 x 2, 8 waves/block -> 16 col tiles
    k_gemm    <<<ggrid, 256, 0, stream>>>(agg2, W2, b2, h2);

    k_pool    <<<N_NODES, 256, 0, stream>>>(h2, batch, gsum, gcnt);
    k_mlp     <<<N_GRAPHS, 256, 0, stream>>>(gsum, gcnt, W3, b3, W4, b4, W5, b5,
                                             W6, b6, W7, b7, (float*)d_out);
}